// LocalGate_67095979098581
// MI455X (gfx1250) — compile-verified
//
#include <hip/hip_runtime.h>
#include <hip/hip_bf16.h>

typedef float v2f __attribute__((ext_vector_type(2)));
typedef float v4f __attribute__((ext_vector_type(4)));
typedef float v8f __attribute__((ext_vector_type(8)));

#define N_TOKENS 16384
#define MODEL_DIM 2048
#define NUM_EXPERTS 64
#define LDS_STRIDE 20           // 16 tokens padded to 20 floats (16B aligned, conflict-free)
#define WAVE_LDS (NUM_EXPERTS * LDS_STRIDE + 16)

#define WMMA_F32(acc, a, b) \
    acc = __builtin_amdgcn_wmma_f32_16x16x4_f32(false, a, false, b, (short)0, acc, false, false)

// ---------------------------------------------------------------------------
// Kernel 1: gate GEMM (fp32 WMMA) + softmax + top-2 + combine weights + probs
// One wave handles a 16-token tile against all 64 experts.
// ---------------------------------------------------------------------------
__global__ __launch_bounds__(128) void gate_kernel(
    const float* __restrict__ X,      // [N, D]
    const float* __restrict__ W,      // [E, D]
    float* __restrict__ probs,        // [N, E]
    float* __restrict__ combine,      // [N*2]
    int* __restrict__ flat_idx)       // [N*2] expert ids
{
    __shared__ float lds[4 * WAVE_LDS];

    const int lane = threadIdx.x & 31;
    const int wave = threadIdx.x >> 5;
    const int tile = blockIdx.x * 4 + wave;   // 16-token tile id
    const int half = lane >> 4;               // 0 or 1
    const int lo   = lane & 15;

    float* wlds = lds + wave * WAVE_LDS;      // [64][LDS_STRIDE] + 16 inv-sums
    float* sums = wlds + NUM_EXPERTS * LDS_STRIDE;

    // Per-lane operand base pointers (A: 16x4 fp32 tile, B: 4x16 fp32 tile).
    // K-slot <-> physical-d mapping: slot(2*half + j) <-> d(k + 4*half + j),
    // identical for A and B, so accumulation over K is exact.
    const float* arow  = X + (size_t)(tile * 16 + lo) * MODEL_DIM + half * 4;
    const float* wrow  = W + (size_t)lo * MODEL_DIM + half * 4;

    v8f acc0 = {}, acc1 = {}, acc2 = {}, acc3 = {};

    #pragma unroll 4
    for (int k = 0; k < MODEL_DIM; k += 8) {
        v4f a  = *(const v4f*)(arow + k);
        v4f b0 = *(const v4f*)(wrow + k);
        v4f b1 = *(const v4f*)(wrow + 16 * MODEL_DIM + k);
        v4f b2 = *(const v4f*)(wrow + 32 * MODEL_DIM + k);
        v4f b3 = *(const v4f*)(wrow + 48 * MODEL_DIM + k);

        v2f al = __builtin_shufflevector(a, a, 0, 1);
        v2f ah = __builtin_shufflevector(a, a, 2, 3);

        WMMA_F32(acc0, al, __builtin_shufflevector(b0, b0, 0, 1));
        WMMA_F32(acc0, ah, __builtin_shufflevector(b0, b0, 2, 3));
        WMMA_F32(acc1, al, __builtin_shufflevector(b1, b1, 0, 1));
        WMMA_F32(acc1, ah, __builtin_shufflevector(b1, b1, 2, 3));
        WMMA_F32(acc2, al, __builtin_shufflevector(b2, b2, 0, 1));
        WMMA_F32(acc2, ah, __builtin_shufflevector(b2, b2, 2, 3));
        WMMA_F32(acc3, al, __builtin_shufflevector(b3, b3, 0, 1));
        WMMA_F32(acc3, ah, __builtin_shufflevector(b3, b3, 2, 3));
    }

    // Spill logits to LDS as [expert][token]:
    // C/D layout: VGPR j, lanes0-15 -> token j, lanes16-31 -> token 8+j; N = lane&15.
    {
        int base0 = (0 * 16 + lo) * LDS_STRIDE + 8 * half;
        int base1 = (1 * 16 + lo) * LDS_STRIDE + 8 * half;
        int base2 = (2 * 16 + lo) * LDS_STRIDE + 8 * half;
        int base3 = (3 * 16 + lo) * LDS_STRIDE + 8 * half;
        *(v4f*)(wlds + base0)     = __builtin_shufflevector(acc0, acc0, 0, 1, 2, 3);
        *(v4f*)(wlds + base0 + 4) = __builtin_shufflevector(acc0, acc0, 4, 5, 6, 7);
        *(v4f*)(wlds + base1)     = __builtin_shufflevector(acc1, acc1, 0, 1, 2, 3);
        *(v4f*)(wlds + base1 + 4) = __builtin_shufflevector(acc1, acc1, 4, 5, 6, 7);
        *(v4f*)(wlds + base2)     = __builtin_shufflevector(acc2, acc2, 0, 1, 2, 3);
        *(v4f*)(wlds + base2 + 4) = __builtin_shufflevector(acc2, acc2, 4, 5, 6, 7);
        *(v4f*)(wlds + base3)     = __builtin_shufflevector(acc3, acc3, 0, 1, 2, 3);
        *(v4f*)(wlds + base3 + 4) = __builtin_shufflevector(acc3, acc3, 4, 5, 6, 7);
    }
    asm volatile("s_wait_dscnt 0x0" ::: "memory");

    // Softmax + top-2 per token: lanes 0..15 each own one token.
    if (lane < 16) {
        const int m = lane;
        float mx = -3.402823466e38f;
        #pragma unroll 8
        for (int e = 0; e < NUM_EXPERTS; ++e)
            mx = fmaxf(mx, wlds[e * LDS_STRIDE + m]);

        float sum = 0.0f;
        float m1 = -1.0f, m2 = -1.0f;
        int i1 = 0, i2 = 0;
        for (int e = 0; e < NUM_EXPERTS; ++e) {
            float v = __expf(wlds[e * LDS_STRIDE + m] - mx);
            wlds[e * LDS_STRIDE + m] = v;     // stash unnormalized prob
            sum += v;
            if (v > m1)      { m2 = m1; i2 = i1; m1 = v; i1 = e; }
            else if (v > m2) { m2 = v;  i2 = e; }
        }
        float inv = 1.0f / sum;
        sums[m] = inv;

        float p1 = m1 * inv, p2 = m2 * inv;
        float w1 = 1.0f / (1.0f + __expf(p2 - p1));   // softmax over top-2 probs
        float w2 = 1.0f - w1;

        int tok = tile * 16 + m;
        combine[2 * tok]     = w1;
        combine[2 * tok + 1] = w2;
        flat_idx[2 * tok]     = i1;
        flat_idx[2 * tok + 1] = i2;
    }
    asm volatile("s_wait_dscnt 0x0" ::: "memory");

    // Coalesced probs write: 16 tokens x 64 experts = 256 float4, all 32 lanes.
    float* pout = probs + (size_t)tile * 16 * NUM_EXPERTS;
    #pragma unroll
    for (int it = 0; it < 8; ++it) {
        int f  = it * 32 + lane;      // float4 index within tile
        int m  = f >> 4;              // token 0..15
        int e0 = (f & 15) * 4;        // first expert of this float4
        float inv = sums[m];
        v4f v;
        v.x = wlds[(e0 + 0) * LDS_STRIDE + m] * inv;
        v.y = wlds[(e0 + 1) * LDS_STRIDE + m] * inv;
        v.z = wlds[(e0 + 2) * LDS_STRIDE + m] * inv;
        v.w = wlds[(e0 + 3) * LDS_STRIDE + m] * inv;
        *(v4f*)(pout + f * 4) = v;
    }
}

// ---------------------------------------------------------------------------
// Kernel 2: per-chunk expert histograms (64 chunks of 512 items)
// ---------------------------------------------------------------------------
__global__ __launch_bounds__(128) void hist_kernel(const int* __restrict__ flat_idx,
                                                   int* __restrict__ hist)
{
    __shared__ int h[NUM_EXPERTS];
    int t = threadIdx.x;
    if (t < NUM_EXPERTS) h[t] = 0;
    __syncthreads();
    int base = blockIdx.x * 512;
    for (int i = t; i < 512; i += 128)
        atomicAdd(&h[flat_idx[base + i]], 1);
    __syncthreads();
    if (t < NUM_EXPERTS) hist[blockIdx.x * NUM_EXPERTS + t] = h[t];
}

// ---------------------------------------------------------------------------
// Kernel 3: 64x64 prefix scan -> per-chunk base offsets + int64 input_splits
// ---------------------------------------------------------------------------
__global__ __launch_bounds__(64) void scan_kernel(const int* __restrict__ hist,
                                                  int* __restrict__ offs,
                                                  long long* __restrict__ splits)
{
    __shared__ int totals[NUM_EXPERTS];
    int e = threadIdx.x;
    int s = 0;
    for (int b = 0; b < 64; ++b) s += hist[b * NUM_EXPERTS + e];
    totals[e] = s;
    __syncthreads();
    int base = 0;
    for (int i = 0; i < e; ++i) base += totals[i];
    splits[e] = (long long)s;
    int off = base;
    for (int b = 0; b < 64; ++b) {
        offs[b * NUM_EXPERTS + e] = off;
        off += hist[b * NUM_EXPERTS + e];
    }
}

// ---------------------------------------------------------------------------
// Kernel 4: stable rank & scatter (one wave per 512-item chunk)
// token_order[r] = i/2 ; reversed_ordering[i] = r
// ---------------------------------------------------------------------------
__global__ __launch_bounds__(32) void rank_kernel(const int* __restrict__ flat_idx,
                                                  const int* __restrict__ offs,
                                                  int* __restrict__ token_order,
                                                  int* __restrict__ reversed)
{
    __shared__ int off[NUM_EXPERTS];
    int lane = threadIdx.x;
    off[lane]      = offs[blockIdx.x * NUM_EXPERTS + lane];
    off[lane + 32] = offs[blockIdx.x * NUM_EXPERTS + lane + 32];
    __syncthreads();

    int base = blockIdx.x * 512;
    for (int g = 0; g < 16; ++g) {
        int i = base + g * 32 + lane;
        int e = flat_idx[i];
        int before = 0;
        bool last = true;
        #pragma unroll
        for (int j = 0; j < 32; ++j) {
            int ej = __shfl(e, j);
            before += (j < lane && ej == e) ? 1 : 0;
            last = last && !((j > lane) && (ej == e));
        }
        int r = off[e] + before;
        token_order[r] = i >> 1;
        reversed[i] = r;
        __syncthreads();
        if (last) off[e] += before + 1;
        __syncthreads();
    }
}

// ---------------------------------------------------------------------------
extern "C" void kernel_launch(void* const* d_in, const int* in_sizes, int n_in,
                              void* d_out, int out_size, void* d_ws, size_t ws_size,
                              hipStream_t stream)
{
    const float* X = (const float*)d_in[0];   // [16384, 2048] fp32
    const float* W = (const float*)d_in[1];   // [64, 2048] fp32

    // Output layout (flat concat, byte offsets):
    char* out = (char*)d_out;
    int*       token_order = (int*)(out);                          // 32768 i32
    int*       reversed    = (int*)(out + 32768u * 4);             // 32768 i32
    float*     combine     = (float*)(out + 65536u * 4);           // 32768 f32
    long long* splits      = (long long*)(out + 98304u * 4);       // 64 i64
    float*     probs       = (float*)(out + 98304u * 4 + 64 * 8);  // 16384*64 f32

    int* flat_idx = (int*)d_ws;          // 32768 i32
    int* hist     = flat_idx + 32768;    // 64*64 i32
    int* offs     = hist + 4096;         // 64*64 i32

    gate_kernel<<<256, 128, 0, stream>>>(X, W, probs, combine, flat_idx);
    hist_kernel<<<64, 128, 0, stream>>>(flat_idx, hist);
    scan_kernel<<<1, 64, 0, stream>>>(hist, offs, splits);
    rank_kernel<<<64, 32, 0, stream>>>(flat_idx, offs, token_order, reversed);
}